// TopKModifier_51719996178486
// MI455X (gfx1250) — compile-verified
//
#include <hip/hip_runtime.h>
#include <stdint.h>

// Per-row top-K magnitude mask. x: rows x 8192 fp32. K = 819 (10% of 8192).
// One workgroup per row: TDM-DMA the row into LDS (tensor_load_to_lds +
// s_wait_tensorcnt), exact 4-pass radix select of the K-th largest |x| bit
// pattern (half-wave-private LDS histograms), masked nontemporal b128 stores.

#define ROW_LEN   8192
#define TOPK      819
#define BLOCK     256
#define NCOPY     16          // histogram copies: one per half-wave (tid/16)

typedef unsigned int u32x4 __attribute__((ext_vector_type(4)));
typedef int          i32x8 __attribute__((ext_vector_type(8)));
typedef int          i32x4 __attribute__((ext_vector_type(4)));
typedef unsigned int uv4   __attribute__((ext_vector_type(4)));
typedef float        fv4   __attribute__((ext_vector_type(4)));

struct SharedBlob {
  unsigned buf[ROW_LEN];        // raw fp32 bits of the row (32 KB)
  unsigned hist[NCOPY][256];    // half-wave-private histograms (16 KB)
  unsigned tot[256];            // reduced histogram (1 KB)
  unsigned sup[32];             // super-bucket sums (groups of 8)
  unsigned sel_prefix;
  unsigned sel_kneed;
};

__global__ __launch_bounds__(BLOCK)
void TopKModifier_51719996178486_kernel(const float* __restrict__ x,
                                        float* __restrict__ out,
                                        int rows) {
  __shared__ SharedBlob s;
  const int row = blockIdx.x;
  if (row >= rows) return;
  const int tid  = threadIdx.x;
  const int copy = tid >> 4;                 // 0..15, two copies per wave

  const float* rowp = x + (size_t)row * ROW_LEN;

  // ---------------- TDM: async DMA the whole 32 KB row into LDS ------------
  // Tensor instructions ignore EXEC and issue once per wave; only wave 0
  // executes this block (other waves branch around it entirely).
  if ((tid >> 5) == 0) {
    unsigned lds_off = (unsigned)(uintptr_t)(&s.buf[0]);   // flat low 32 = LDS addr
    unsigned long long ga = (unsigned long long)(uintptr_t)rowp;

    // D# group 0 (128b): count=1 | lds_addr | global_addr[56:0] | type=2
    u32x4 g0;
    g0[0] = 1u;
    g0[1] = lds_off;
    g0[2] = (unsigned)(ga & 0xffffffffull);
    g0[3] = (unsigned)((ga >> 32) & 0x01ffffffull) | (2u << 30);

    // D# group 1 (256b): 2D tensor, 1 x 8192 tile of 4-byte elements.
    i32x8 g1;
    g1[0] = (int)(2u << 16);                            // data_size=2 (4B), mask=0
    g1[1] = (int)((unsigned)ROW_LEN << 16);             // tensor_dim0[15:0] @ [63:48]
    g1[2] = (int)(1u << 16);                            // tensor_dim1 = 1
    g1[3] = (int)((unsigned)ROW_LEN << 16);             // tile_dim0 = 8192
    g1[4] = 1;                                          // tile_dim1=1, tile_dim2 unused
    g1[5] = ROW_LEN;                                    // tensor_dim0_stride lo
    g1[6] = (int)(((unsigned)ROW_LEN & 0xffffu) << 16); // stride0 hi=0; stride1 lo
    g1[7] = 0;

    i32x4 gz = {0, 0, 0, 0};
#if __clang_major__ >= 23
    i32x8 gz8 = {0, 0, 0, 0, 0, 0, 0, 0};
    __builtin_amdgcn_tensor_load_to_lds(g0, g1, gz, gz, gz8, 0);
#else
    __builtin_amdgcn_tensor_load_to_lds(g0, g1, gz, gz, 0);
#endif
    __builtin_amdgcn_s_wait_tensorcnt(0);
  }
  __syncthreads();

  // ------------- Exact radix select of K-th largest |x| bit pattern --------
  // Non-negative IEEE floats order identically to their unsigned bit patterns.
  const uv4* bufv = (const uv4*)s.buf;
  unsigned prefix = 0;
  unsigned kneed  = TOPK;

  for (int pass = 0; pass < 4; ++pass) {
    const int shift = 24 - 8 * pass;

    // Clear histogram copies (16*256 entries / 256 threads = 16 each).
    unsigned* hflat = &s.hist[0][0];
    #pragma unroll
    for (int i = tid; i < NCOPY * 256; i += BLOCK) hflat[i] = 0u;
    __syncthreads();

    // Elements whose already-selected high bytes match `prefix` get
    // histogrammed on the current byte. Pass 0 takes everything.
    unsigned himask = 0u;
    if (pass > 0) himask = 0xffffffffu << (shift + 8);   // shift+8 <= 24 here

    unsigned* myhist = s.hist[copy];
    for (int i = tid; i < ROW_LEN / 4; i += BLOCK) {
      uv4 v = bufv[i];
      unsigned a;
      a = v.x & 0x7fffffffu;
      if (((a ^ prefix) & himask) == 0u) atomicAdd(&myhist[(a >> shift) & 0xffu], 1u);
      a = v.y & 0x7fffffffu;
      if (((a ^ prefix) & himask) == 0u) atomicAdd(&myhist[(a >> shift) & 0xffu], 1u);
      a = v.z & 0x7fffffffu;
      if (((a ^ prefix) & himask) == 0u) atomicAdd(&myhist[(a >> shift) & 0xffu], 1u);
      a = v.w & 0x7fffffffu;
      if (((a ^ prefix) & himask) == 0u) atomicAdd(&myhist[(a >> shift) & 0xffu], 1u);
    }
    __syncthreads();

    // Reduce copies: thread t owns bucket t.
    unsigned tot = 0;
    #pragma unroll
    for (int c = 0; c < NCOPY; ++c) tot += s.hist[c][tid];
    s.tot[tid] = tot;
    __syncthreads();

    // Two-level suffix count: sup[g] = sum of tot[8g..8g+7].
    if (tid < 32) {
      unsigned ssum = 0;
      #pragma unroll
      for (int j = 0; j < 8; ++j) ssum += s.tot[tid * 8 + j];
      s.sup[tid] = ssum;
    }
    __syncthreads();

    // above(t) = count strictly above bucket t. The unique t with
    // above < kneed <= above + tot[t] carries the K-th value's byte.
    unsigned above = 0;
    const int g = tid >> 3;
    for (int j = g + 1; j < 32; ++j) above += s.sup[j];
    for (int j = tid + 1; j < (g + 1) * 8; ++j) above += s.tot[j];
    if (above < kneed && above + tot >= kneed) {
      s.sel_prefix = prefix | ((unsigned)tid << shift);
      s.sel_kneed  = kneed - above;
    }
    __syncthreads();
    prefix = s.sel_prefix;
    kneed  = s.sel_kneed;
    __syncthreads();
  }

  // prefix == exact bit pattern of the K-th largest |x| in the row.
  const unsigned thr = prefix;

  // ------------------- Masked write-out (nontemporal b128) -----------------
  fv4* outv = (fv4*)(out + (size_t)row * ROW_LEN);
  for (int i = tid; i < ROW_LEN / 4; i += BLOCK) {
    uv4 v = bufv[i];
    fv4 o;
    o.x = ((v.x & 0x7fffffffu) >= thr) ? __uint_as_float(v.x) : 0.0f;
    o.y = ((v.y & 0x7fffffffu) >= thr) ? __uint_as_float(v.y) : 0.0f;
    o.z = ((v.z & 0x7fffffffu) >= thr) ? __uint_as_float(v.z) : 0.0f;
    o.w = ((v.w & 0x7fffffffu) >= thr) ? __uint_as_float(v.w) : 0.0f;
    __builtin_nontemporal_store(o, &outv[i]);
  }
}

extern "C" void kernel_launch(void* const* d_in, const int* in_sizes, int n_in,
                              void* d_out, int out_size, void* d_ws, size_t ws_size,
                              hipStream_t stream) {
  (void)n_in; (void)out_size; (void)d_ws; (void)ws_size;
  const float* x = (const float*)d_in[0];
  float* out = (float*)d_out;
  const int rows = in_sizes[0] / ROW_LEN;   // 4096 rows of 8192
  TopKModifier_51719996178486_kernel<<<rows, BLOCK, 0, stream>>>(x, out, rows);
}